// GraphSage_Net_13151189860606
// MI455X (gfx1250) — compile-verified
//
#include <hip/hip_runtime.h>
#include <hip/hip_bf16.h>
#include <math.h>

#define N_NODES 50000
#define N_EDGES 800000
#define N_EVAL  500000
#define D 96
#define ROWS_PER_BLOCK 64   // 4 row-tiles of 16

typedef __attribute__((ext_vector_type(2))) float v2f;
typedef __attribute__((ext_vector_type(8))) float v8f;

// ---------------------------------------------------------------------------
// Zero a float buffer (float4 stores; n is a multiple of 4 here).
// ---------------------------------------------------------------------------
__global__ void gs_zero_kernel(float* __restrict__ p, int n) {
    int i4 = (blockIdx.x * blockDim.x + threadIdx.x) * 4;
    if (i4 + 4 <= n) {
        *(float4*)(p + i4) = make_float4(0.f, 0.f, 0.f, 0.f);
    } else {
        for (int j = i4; j < n; ++j) p[j] = 0.f;
    }
}

// ---------------------------------------------------------------------------
// Scatter-add aggregation: aggr[dst[e]] += x[src[e]]   (segment_sum)
// One wave32 per edge; lane l handles features l, l+32, l+64.
// fp32 atomics resolve in L2 (entire 19.2MB aggr buffer is L2-resident).
// ---------------------------------------------------------------------------
__global__ __launch_bounds__(256)
void gs_scatter_add_kernel(const float* __restrict__ x,
                           const int* __restrict__ src,
                           const int* __restrict__ dst,
                           float* __restrict__ aggr, int nedges) {
    int wave = (blockIdx.x * blockDim.x + threadIdx.x) >> 5;
    int lane = threadIdx.x & 31;
    if (wave >= nedges) return;
    int s = src[wave];           // wave-uniform -> scalar load path
    int d = dst[wave];
    const float* xs = x + s * D;
    float* ad = aggr + d * D;
#pragma unroll
    for (int k = 0; k < 3; ++k) {
        int f = lane + 32 * k;
        atomicAdd(&ad[f], xs[f]);
    }
}

// ---------------------------------------------------------------------------
// Fused SAGE linear:  out = act( Xa @ Wa + Xb @ Wb + bias )
// Native fp32 WMMA (V_WMMA_F32_16X16X4_F32).
//  - Block = 192 threads (6 waves); wave w owns output columns [16w, 16w+16).
//  - Block covers 64 rows: Xa/Xb tiles staged in LDS once (48 KB), shared by
//    all 6 waves; A-fragments then come from LDS (ds_load_b64).
//  - Each wave keeps 4 accumulators (rows m0+0/16/32/48) so every strided
//    B-fragment (weight) load is amortized over 4 WMMAs.
// Fragment layouts per CDNA5 ISA 7.12.2:
//   A(m,k): lane = (m&15) + 16*(k>=2), vgpr = k&1
//   B(k,n): lane = (n&15) + 16*(k>=2), vgpr = k&1
//   C vgpr r: m = r + 8*(lane>=16), n = lane&15
// ---------------------------------------------------------------------------
__global__ __launch_bounds__(192)
void gs_gemm_kernel(const float* __restrict__ Xa, const float* __restrict__ Wa,
                    const float* __restrict__ Xb, const float* __restrict__ Wb,
                    const float* __restrict__ bias, float* __restrict__ out,
                    int do_relu) {
    __shared__ float lA[ROWS_PER_BLOCK * D];   // 24 KB
    __shared__ float lB[ROWS_PER_BLOCK * D];   // 24 KB

    const int tid  = threadIdx.x;
    const int lane = tid & 31;
    const int wv   = tid >> 5;                 // 0..5 -> column tile
    const int m0   = blockIdx.x * ROWS_PER_BLOCK;

    // Cooperative LDS fill: 64x96 floats per matrix = 1536 float4s, 192 thr
    // -> 8 float4 per thread per matrix. Clamp tail rows (reads only).
#pragma unroll
    for (int it = 0; it < (ROWS_PER_BLOCK * D / 4) / 192; ++it) {
        int i  = (it * 192 + tid) * 4;
        int r  = i / D;
        int cc = i % D;                         // D % 4 == 0: float4 stays in-row
        int gr = m0 + r; if (gr > N_NODES - 1) gr = N_NODES - 1;
        *(float4*)&lA[i] = *(const float4*)&Xa[gr * D + cc];
        *(float4*)&lB[i] = *(const float4*)&Xb[gr * D + cc];
    }
    __syncthreads();

    const int n0    = wv * 16;
    const int kHalf = (lane >> 4) << 1;        // 0 or 2
    const int nB    = n0 + (lane & 15);        // weight column for this lane
    const int rA    = (lane & 15);             // row within each 16-row tile

    v8f acc0 = {0.f,0.f,0.f,0.f,0.f,0.f,0.f,0.f};
    v8f acc1 = acc0, acc2 = acc0, acc3 = acc0;

#pragma unroll
    for (int k0 = 0; k0 < D; k0 += 4) {
        const int k = k0 + kHalf;
        // B-fragments (strided weight reads, WGP$/L2 resident), loaded once,
        // used by 4 WMMAs each.
        v2f ba, bb;
        ba.x = Wa[k * D + nB];  ba.y = Wa[(k + 1) * D + nB];
        bb.x = Wb[k * D + nB];  bb.y = Wb[(k + 1) * D + nB];

#define GS_TILE(T, ACC)                                                        \
        {                                                                      \
            v2f a = *(const v2f*)&lA[(rA + (T) * 16) * D + k];                 \
            ACC = __builtin_amdgcn_wmma_f32_16x16x4_f32(                       \
                      false, a, false, ba, (short)0, ACC, false, false);       \
            v2f b = *(const v2f*)&lB[(rA + (T) * 16) * D + k];                 \
            ACC = __builtin_amdgcn_wmma_f32_16x16x4_f32(                       \
                      false, b, false, bb, (short)0, ACC, false, false);       \
        }
        GS_TILE(0, acc0)
        GS_TILE(1, acc1)
        GS_TILE(2, acc2)
        GS_TILE(3, acc3)
#undef GS_TILE
    }

    const float bv    = bias[nB];
    const int   mBase = m0 + ((lane >> 4) ? 8 : 0);

#define GS_STORE(T, ACC)                                                       \
    {                                                                          \
        _Pragma("unroll")                                                      \
        for (int r = 0; r < 8; ++r) {                                          \
            int m = mBase + (T) * 16 + r;                                      \
            if (m < N_NODES) {                                                 \
                float v = ACC[r] + bv;                                         \
                if (do_relu) v = fmaxf(v, 0.f);                                \
                out[m * D + nB] = v;                                           \
            }                                                                  \
        }                                                                      \
    }
    GS_STORE(0, acc0)
    GS_STORE(1, acc1)
    GS_STORE(2, acc2)
    GS_STORE(3, acc3)
#undef GS_STORE
}

// ---------------------------------------------------------------------------
// Edge scoring: sigmoid( dot(h[E0[e]], h[E1[e]]) ), one wave32 per edge.
// ---------------------------------------------------------------------------
__global__ __launch_bounds__(256)
void gs_edge_score_kernel(const float* __restrict__ h,
                          const int* __restrict__ e0,
                          const int* __restrict__ e1,
                          float* __restrict__ out, int n) {
    int wave = (blockIdx.x * blockDim.x + threadIdx.x) >> 5;
    int lane = threadIdx.x & 31;
    if (wave >= n) return;
    const float* a = h + e0[wave] * D;
    const float* b = h + e1[wave] * D;
    float acc = 0.f;
#pragma unroll
    for (int k = 0; k < 3; ++k) {
        int f = lane + 32 * k;
        acc = fmaf(a[f], b[f], acc);
    }
#pragma unroll
    for (int off = 16; off > 0; off >>= 1)
        acc += __shfl_xor(acc, off, 32);      // wave32 butterfly
    if (lane == 0)
        out[wave] = 1.0f / (1.0f + __expf(-acc));
}

// ---------------------------------------------------------------------------
extern "C" void kernel_launch(void* const* d_in, const int* in_sizes, int n_in,
                              void* d_out, int out_size, void* d_ws, size_t ws_size,
                              hipStream_t stream) {
    const float* Features = (const float*)d_in[0];
    const int*   A        = (const int*)d_in[1];   // [2, N_EDGES]: src row, dst row
    const int*   E        = (const int*)d_in[2];   // [2, N_EVAL]
    const float* W1l      = (const float*)d_in[3];
    const float* W1r      = (const float*)d_in[4];
    const float* b1       = (const float*)d_in[5];
    const float* W2l      = (const float*)d_in[6];
    const float* W2r      = (const float*)d_in[7];
    const float* b2       = (const float*)d_in[8];
    float*       out      = (float*)d_out;

    float* aggr = (float*)d_ws;                       // 19.2 MB, reused per layer
    float* h    = aggr + (size_t)N_NODES * D;         // 19.2 MB
    float* h2   = h    + (size_t)N_NODES * D;         // 19.2 MB

    const int NF    = N_NODES * D;
    const int zgrid = (NF / 4 + 255) / 256;
    const int sgrid = (N_EDGES + 7) / 8;              // 8 waves/block, 1 edge/wave
    const int ggrid = (N_NODES + ROWS_PER_BLOCK - 1) / ROWS_PER_BLOCK;  // 782
    const int egrid = (N_EVAL + 7) / 8;

    // Layer 1: aggr = segsum(Features); h = relu(aggr@W1l + Features@W1r + b1)
    gs_zero_kernel<<<zgrid, 256, 0, stream>>>(aggr, NF);
    gs_scatter_add_kernel<<<sgrid, 256, 0, stream>>>(Features, A, A + N_EDGES, aggr, N_EDGES);
    gs_gemm_kernel<<<ggrid, 192, 0, stream>>>(aggr, W1l, Features, W1r, b1, h, 1);

    // Layer 2: aggr = segsum(h); h2 = aggr@W2l + h@W2r + b2
    gs_zero_kernel<<<zgrid, 256, 0, stream>>>(aggr, NF);
    gs_scatter_add_kernel<<<sgrid, 256, 0, stream>>>(h, A, A + N_EDGES, aggr, N_EDGES);
    gs_gemm_kernel<<<ggrid, 192, 0, stream>>>(aggr, W2l, h, W2r, b2, h2, 0);

    // Edge scoring
    gs_edge_score_kernel<<<egrid, 256, 0, stream>>>(h2, E, E + N_EVAL, out, N_EVAL);
}